// ConvIntrinsicLite_52578989637715
// MI455X (gfx1250) — compile-verified
//
#include <hip/hip_runtime.h>

typedef __attribute__((ext_vector_type(2))) float v2f;
typedef __attribute__((ext_vector_type(8))) float v8f;

#define NVERT 25000
#define RA    40
#define FDIM  32
#define ODIM  64
#define NROT  8
// bary row stride = RA*3*2 = 240 floats; out row stride = NROT*ODIM = 512 floats

// ---------------------------------------------------------------------------
// Setup: csum[k] = sum_{r,a} interp_coef[r,a,k]   (40 floats)
//        bfrag   = kernel_weights pre-swizzled into WMMA B-fragment lane order
//                  (8 tiles x 8 K-steps x 32 lanes x 2 floats = 4096 floats)
// B layout for V_WMMA_F32_16X16X4_F32 (4x16 B, mirrors documented 16x4 A):
//   lanes 0-15 : v0 = B[K=4s+0][N=L], v1 = B[K=4s+1][N=L]
//   lanes16-31 : v0 = B[K=4s+2][N=L-16], v1 = B[K=4s+3][N=L-16]
// with B[f][n'] = W[ko][f], ko = 16*tile + n'
// ---------------------------------------------------------------------------
__global__ __launch_bounds__(256)
void setup_kernel(const float* __restrict__ coef, const float* __restrict__ kw,
                  float* __restrict__ csum, float* __restrict__ bfrag) {
  int t = blockIdx.x * blockDim.x + threadIdx.x;
  if (t < RA) {
    float s = 0.f;
    for (int m = 0; m < RA; ++m) s += coef[m * RA + t];
    csum[t] = s;
  }
  if (t < 4096) {
    int h    = t & 1;
    int L    = (t >> 1) & 31;
    int ts   = t >> 6;      // tile*8 + s
    int s    = ts & 7;
    int tile = ts >> 3;
    int ko, f;
    if (L < 16) { ko = 16 * tile + L;      f = 4 * s + h;     }
    else        { ko = 16 * tile + L - 16; f = 4 * s + 2 + h; }
    bfrag[t] = kw[ko * FDIM + f];
  }
}

// ---------------------------------------------------------------------------
// Main kernel: 8 waves/block, each wave computes 16 output rows.
// ---------------------------------------------------------------------------
__global__ __launch_bounds__(256)
void conv_main(const float* __restrict__ mesh, const float* __restrict__ bary,
               const float* __restrict__ bias, const float* __restrict__ csum,
               const float* __restrict__ bfrag, float* __restrict__ out) {
  __shared__ float s_csum[RA];
  __shared__ float s_S[8][16 * 34];   // stride-34 padding: conflict-free transpose

  const int tid  = threadIdx.x;
  const int lane = tid & 31;
  const int wave = tid >> 5;

  if (tid < RA) s_csum[tid] = csum[tid];
  __syncthreads();

  const int group = blockIdx.x * 8 + wave;   // 16-row group index
  const int n0    = group * 16;
  float* Srow = &s_S[wave][0];

  // ---- gather + interp-collapse phase: lane == feature f ----
  for (int r = 0; r < 16; ++r) {
    int n = n0 + r;
    if (n >= NVERT) n = NVERT - 1;           // clamp; garbage rows never stored
    const float2* b2 = (const float2*)(bary + (size_t)n * 240);
    float2 p0 = b2[lane];
    float2 p1 = b2[lane + 32];
    float2 p2 = b2[lane + 64];
    float2 p3 = (lane < 24) ? b2[lane + 96] : make_float2(0.f, 0.f);

    int   i0 = (int)p0.x, i1 = (int)p1.x, i2 = (int)p2.x, i3 = (int)p3.x;
    float w0 = p0.y * s_csum[lane / 3];
    float w1 = p1.y * s_csum[(lane + 32) / 3];
    float w2 = p2.y * s_csum[(lane + 64) / 3];
    float w3 = (lane < 24) ? p3.y * s_csum[(lane + 96) / 3] : 0.f;

    float accA = 0.f, accB = 0.f;
#define GATHER_SLOT(IV, WV, CNT)                                               \
    _Pragma("unroll")                                                          \
    for (int pp = 0; pp < (CNT); pp += 2) {                                    \
      int   ida = __builtin_amdgcn_readlane((IV), pp);                         \
      float wa  = __uint_as_float(                                             \
                    __builtin_amdgcn_readlane(__float_as_uint(WV), pp));       \
      int   idb = __builtin_amdgcn_readlane((IV), pp + 1);                     \
      float wb  = __uint_as_float(                                             \
                    __builtin_amdgcn_readlane(__float_as_uint(WV), pp + 1));   \
      accA = fmaf(wa, mesh[(size_t)ida * FDIM + lane], accA);                  \
      accB = fmaf(wb, mesh[(size_t)idb * FDIM + lane], accB);                  \
    }
    GATHER_SLOT(i0, w0, 32)
    GATHER_SLOT(i1, w1, 32)
    GATHER_SLOT(i2, w2, 32)
    GATHER_SLOT(i3, w3, 24)
#undef GATHER_SLOT
    Srow[r * 34 + lane] = accA + accB;       // S[row][f]
  }
  // same-wave LDS RAW: DScnt handled by compiler, no barrier needed

  // ---- A fragments (documented 32-bit 16x4 layout) ----
  const int half = lane >> 4;
  const int mrow = lane & 15;
  v2f a[8];
#pragma unroll
  for (int s = 0; s < 8; ++s) {
    int f = 4 * s + 2 * half;
    a[s][0] = Srow[mrow * 34 + f];
    a[s][1] = Srow[mrow * 34 + f + 1];
  }

  const float2* bf2 = (const float2*)bfrag;
  const int col = lane & 15;

#pragma unroll
  for (int t = 0; t < 4; ++t) {
    v8f c0 = {0.f, 0.f, 0.f, 0.f, 0.f, 0.f, 0.f, 0.f};
    v8f c1 = c0;
#pragma unroll
    for (int s = 0; s < 8; ++s) {
      float2 bl0 = bf2[(t * 8 + s) * 32 + lane];        // k=0 tile
      float2 bl1 = bf2[((t + 4) * 8 + s) * 32 + lane];  // k=1 tile
      v2f b0; b0[0] = bl0.x; b0[1] = bl0.y;
      v2f b1; b1[0] = bl1.x; b1[1] = bl1.y;
      c0 = __builtin_amdgcn_wmma_f32_16x16x4_f32(false, a[s], false, b0,
                                                 (short)0, c0, false, false);
      c1 = __builtin_amdgcn_wmma_f32_16x16x4_f32(false, a[s], false, b1,
                                                 (short)0, c1, false, false);
    }
    float bs0 = bias[16 * t + col];          // k=0 bias
    float bs1 = bias[ODIM + 16 * t + col];   // k=1 bias
#pragma unroll
    for (int i = 0; i < 8; ++i) {
      float z0 = c0[i] + bs0; z0 = z0 > 0.f ? z0 : 0.f;
      float z1 = c1[i] + bs1; z1 = z1 > 0.f ? z1 : 0.f;
      float v  = z0 + z1;
      int n = n0 + i + 8 * half;             // C/D layout: M = vgpr + 8*(lane/16)
      if (n < NVERT) {
        float* op = out + (size_t)n * (NROT * ODIM) + 16 * t + col;
#pragma unroll
        for (int sr = 0; sr < NROT; ++sr) op[sr * ODIM] = v;  // roll-invariant
      }
    }
  }
}

// ---------------------------------------------------------------------------
extern "C" void kernel_launch(void* const* d_in, const int* in_sizes, int n_in,
                              void* d_out, int out_size, void* d_ws, size_t ws_size,
                              hipStream_t stream) {
  const float* mesh = (const float*)d_in[0];  // (N,F)
  const float* bary = (const float*)d_in[1];  // (N,R,A,3,2)
  const float* coef = (const float*)d_in[2];  // (R,A,RA)
  const float* kw   = (const float*)d_in[3];  // (K,O,F)
  const float* bias = (const float*)d_in[4];  // (K,O)
  float* out = (float*)d_out;

  float* csum  = (float*)d_ws;                // 40 floats
  float* bfrag = (float*)d_ws + 64;           // 4096 floats (8B aligned)

  setup_kernel<<<16, 256, 0, stream>>>(coef, kw, csum, bfrag);

  int groups = (NVERT + 15) / 16;
  int blocks = (groups + 7) / 8;
  conv_main<<<blocks, 256, 0, stream>>>(mesh, bary, bias, csum, bfrag, out);
}